// GaussianMomentDescriptorT_89567247991295
// MI455X (gfx1250) — compile-verified
//
#include <hip/hip_runtime.h>

#define N_RADIAL 5
#define N_BASIS  7
#define N_SPECIES 119
#define N_ATOMS  25000
#define N_EDGES  1000000
#define ACC_STRIDE 200   // full layout: m0(5) + m1(15) + m2(45) + m3(135)
#define CMP_SIZE   100   // compact: m0(5) + m1(15) + m2u(5*6) + m3u(5*10)
#define OUT_STRIDE 360   // 5+15+15+15+35+75+75+125
#define WAVES_PER_BLOCK 8

typedef float v2f __attribute__((ext_vector_type(2)));
typedef float v8f __attribute__((ext_vector_type(8)));

__device__ __forceinline__ void atomic_add_f32(float* p, float v) {
  // relaxed agent-scope fadd -> native global_atomic_add_f32 on gfx1250 (L2 atomic)
  __hip_atomic_fetch_add(p, v, __ATOMIC_RELAXED, __HIP_MEMORY_SCOPE_AGENT);
}

__global__ void zero_ws_kernel(float* __restrict__ ws, int n) {
  for (int i = blockIdx.x * blockDim.x + threadIdx.x; i < n;
       i += gridDim.x * blockDim.x)
    ws[i] = 0.0f;
}

// ---------------- Phase 1: per-edge moments, symmetric-compact scatter-add.
// m2/m3 are fully symmetric in the Cartesian indices -> accumulate only the
// 6 (pairs) / 10 (triples) unique components: 100 atomics/edge instead of 200.
__global__ void edge_kernel(const float* __restrict__ dr_vec,
                            const int*   __restrict__ Z,
                            const int*   __restrict__ nbr,   // (2, N_EDGES)
                            const float* __restrict__ coeffs,
                            float*       __restrict__ ws) {
  int e = blockIdx.x * blockDim.x + threadIdx.x;
  if (e >= N_EDGES) return;

  float x = dr_vec[3 * e + 0];
  float y = dr_vec[3 * e + 1];
  float z = dr_vec[3 * e + 2];
  float dr  = sqrtf(x * x + y * y + z * z);
  float inv = 1.0f / (dr + 1e-5f);
  float dn[3] = {x * inv, y * inv, z * inv};

  const float betta    = 49.0f / 36.0f;          // N_BASIS^2 / R_MAX^2
  const float rad_norm = 0.8981633f;             // (2*betta/pi)^0.75
  const float inv_s7   = 0.37796447f;            // 1/sqrt(7)
  float cutoff = (dr < 6.0f)
                   ? 0.5f * (__cosf(0.5235987755982988f * dr) + 1.0f)  // pi/6
                   : 0.0f;

  int zi = Z[nbr[e]];
  int aj = nbr[N_EDGES + e];
  int zj = Z[aj];
  const float* c = coeffs + ((long)(zi * N_SPECIES + zj) * N_RADIAL * N_BASIS);
  __builtin_prefetch(c, 0, 0);                   // global_prefetch_b8

  float basis[N_BASIS];
#pragma unroll
  for (int b = 0; b < N_BASIS; ++b) {            // shifts = 0,1,...,6
    float d = dr - (float)b;
    basis[b] = rad_norm * __expf(-betta * d * d);
  }

  float rad[N_RADIAL];
#pragma unroll
  for (int r = 0; r < N_RADIAL; ++r) {
    float acc = 0.0f;
#pragma unroll
    for (int b = 0; b < N_BASIS; ++b) acc += basis[b] * c[r * N_BASIS + b];
    rad[r] = acc * cutoff * inv_s7;
  }

  // unique symmetric direction products
  float q2[6], q3[10];
  {
    int p = 0;
#pragma unroll
    for (int i = 0; i < 3; ++i)
#pragma unroll
      for (int j = i; j < 3; ++j) q2[p++] = dn[i] * dn[j];
    p = 0;
#pragma unroll
    for (int i = 0; i < 3; ++i)
#pragma unroll
      for (int j = i; j < 3; ++j)
#pragma unroll
        for (int k = j; k < 3; ++k) q3[p++] = dn[i] * dn[j] * dn[k];
  }

  float* acc = ws + (long)aj * ACC_STRIDE;
  for (int r = 0; r < N_RADIAL; ++r) {
    atomic_add_f32(acc + r, rad[r]);
    for (int i = 0; i < 3; ++i)
      atomic_add_f32(acc + 5 + r * 3 + i, rad[r] * dn[i]);
    for (int u = 0; u < 6; ++u)
      atomic_add_f32(acc + 20 + r * 6 + u, rad[r] * q2[u]);
    for (int u = 0; u < 10; ++u)
      atomic_add_f32(acc + 50 + r * 10 + u, rad[r] * q3[u]);
  }
}

// ---------------- Phase 1b: in-place expansion compact(100) -> full(200).
__device__ __forceinline__ int u2idx(int i, int j) {
  int a = min(i, j), b = max(i, j);
  return (a == 0) ? b : ((a == 1) ? 2 + b : 5);
}

__global__ void expand_kernel(float* __restrict__ ws) {
  // sorted-triple LUT: index a*9+b*3+c (a<=b<=c) -> unique id 0..9
  const int lut3[27] = {0, 1, 2, -1, 3, 4, -1, -1, 5, -1, -1, -1, -1, 6,
                        7, -1, -1, 8, -1, -1, -1, -1, -1, -1, -1, -1, 9};
  int a = blockIdx.x * blockDim.x + threadIdx.x;
  if (a >= N_ATOMS) return;
  float* m = ws + (long)a * ACC_STRIDE;
  float c2[30], c3[50];
  for (int i = 0; i < 30; ++i) c2[i] = m[20 + i];
  for (int i = 0; i < 50; ++i) c3[i] = m[50 + i];
  // full m2 @20 (45), full m3 @65 (135)
  for (int r = 0; r < 5; ++r)
    for (int i = 0; i < 3; ++i)
      for (int j = 0; j < 3; ++j)
        m[20 + r * 9 + i * 3 + j] = c2[r * 6 + u2idx(i, j)];
  for (int r = 0; r < 5; ++r)
    for (int i = 0; i < 3; ++i)
      for (int j = 0; j < 3; ++j)
        for (int k = 0; k < 3; ++k) {
          int s0 = min(i, min(j, k));
          int s2 = max(i, max(j, k));
          int s1 = i + j + k - s0 - s2;
          m[65 + r * 27 + i * 9 + j * 3 + k] =
              c3[r * 10 + lut3[s0 * 9 + s1 * 3 + s2]];
        }
}

// ---------------- Phase 2: batched 5xK Gram matrices via V_WMMA_F32_16X16X4_F32
// A tile: rows 5t..5t+4 hold atom (aBase+t)'s m-block (t=0..2), row 15 = 0.
// ISA 16x4 f32 A layout: element (M,K) -> lane = M + 16*(K>>1), vgpr = K&1.
// B = A^T with the mirrored N-across-lanes layout => B fragment == A fragment.
// Fragments come from LDS (ds_load) with multiply-masking: no EXEC branching.
__device__ __forceinline__ void gram_wmma(const float* __restrict__ sm,
                                          int mOff, int K, int nSteps,
                                          float rowMask, int r, int half,
                                          int col, int aBase, int outBase,
                                          float* __restrict__ out) {
  v8f cacc = {};
  for (int kk = 0; kk < nSteps * 4; kk += 4) {
    int k0 = kk + half * 2;
    int k1 = k0 + 1;
    float a0 = sm[mOff + r * K + min(k0, K - 1)] *
               ((k0 < K) ? rowMask : 0.0f);
    float a1 = sm[mOff + r * K + min(k1, K - 1)] *
               ((k1 < K) ? rowMask : 0.0f);
    v2f a = {a0, a1};
    cacc = __builtin_amdgcn_wmma_f32_16x16x4_f32(
        /*neg_a=*/false, a, /*neg_b=*/false, a,
        /*c_mod=*/(short)0, cacc, /*reuse_a=*/false, /*reuse_b=*/false);
  }
  // D layout: lane l, vgpr v holds D[v + 8*(l>>4)][l&15]
  int nT = col / 5, ns = col % 5;
#pragma unroll
  for (int v = 0; v < 8; ++v) {
    int M  = v + 8 * half;
    int mT = M / 5, mr = M % 5;
    if (mT == nT && mT < 3 && ns <= mr) {          // lower triangle, same atom
      int atom = aBase + mT;
      if (atom < N_ATOMS)
        out[(long)atom * OUT_STRIDE + outBase + (mr * (mr + 1)) / 2 + ns] =
            cacc[v];
    }
  }
}

__global__ void gram_kernel(const float* __restrict__ ws,
                            float* __restrict__ out) {
  __shared__ float smem[WAVES_PER_BLOCK * 3 * ACC_STRIDE];  // 19.2 KB / WGP blk
  int lane  = threadIdx.x & 31;
  int wIB   = threadIdx.x >> 5;
  int wave  = blockIdx.x * WAVES_PER_BLOCK + wIB;
  int aBase = wave * 3;
  float* sm = smem + wIB * (3 * ACC_STRIDE);

  // cooperative coalesced stage: 3 atoms x 200 floats per wave (clamped)
  for (int idx = lane; idx < 3 * ACC_STRIDE; idx += 32) {
    int atom = min(aBase + idx / ACC_STRIDE, N_ATOMS - 1);
    sm[idx] = ws[(long)atom * ACC_STRIDE + (idx % ACC_STRIDE)];
  }
  __syncthreads();   // all threads reach this: no early returns in this kernel

  int row  = lane & 15;           // M for A-frag, N for D-frag
  int half = lane >> 4;
  int t = row / 5;                // t==3 -> padding row 15
  int r = row % 5;
  float rowMask = (t < 3 && (aBase + t) < N_ATOMS) ? 1.0f : 0.0f;
  const float* mrow = sm + min(t, 2) * ACC_STRIDE;

  gram_wmma(mrow,  5,  3, 1, rowMask, r, half, row, aBase,  5, out); // contr_1
  gram_wmma(mrow, 20,  9, 3, rowMask, r, half, row, aBase, 20, out); // contr_2
  gram_wmma(mrow, 65, 27, 7, rowMask, r, half, row, aBase, 35, out); // contr_3
}

// ---------------- Phase 3: small triple contractions (scalar, one thread/atom)
__global__ void contract_kernel(const float* __restrict__ ws,
                                float* __restrict__ out) {
  int a = blockIdx.x * blockDim.x + threadIdx.x;
  if (a >= N_ATOMS) return;
  const float* m = ws + (long)a * ACC_STRIDE;
  float m1[15], m2[45], m3[135];
  for (int i = 0; i < 15; ++i)  m1[i] = m[5 + i];
  for (int i = 0; i < 45; ++i)  m2[i] = m[20 + i];
  for (int i = 0; i < 135; ++i) m3[i] = m[65 + i];
  float* o = out + (long)a * OUT_STRIDE;

  for (int r = 0; r < 5; ++r) o[r] = m[r];                  // contr_0 @0
  int p = 0;                                                // contr_4 @50
  for (int r = 0; r < 5; ++r)
    for (int s = 0; s <= r; ++s)
      for (int t = 0; t <= s; ++t) {
        float acc = 0.f;
        for (int i = 0; i < 3; ++i)
          for (int j = 0; j < 3; ++j)
            for (int k = 0; k < 3; ++k)
              acc += m2[r*9+i*3+j] * m2[s*9+i*3+k] * m2[t*9+j*3+k];
        o[50 + p++] = acc;
      }
  p = 0;                                                    // contr_5 @85
  for (int r = 0; r < 5; ++r)
    for (int s = 0; s <= r; ++s)
      for (int t = 0; t < 5; ++t) {
        float acc = 0.f;
        for (int i = 0; i < 3; ++i)
          for (int j = 0; j < 3; ++j)
            acc += m1[r*3+i] * m1[s*3+j] * m2[t*9+i*3+j];
        o[85 + p++] = acc;
      }
  p = 0;                                                    // contr_6 @160
  for (int r = 0; r < 5; ++r)
    for (int s = 0; s <= r; ++s) {
      float T[9] = {0.f,0.f,0.f,0.f,0.f,0.f,0.f,0.f,0.f};
      for (int i = 0; i < 3; ++i)
        for (int j = 0; j < 3; ++j)
          for (int k = 0; k < 3; ++k)
            for (int l = 0; l < 3; ++l)
              T[k*3+l] += m3[r*27+i*9+j*3+k] * m3[s*27+i*9+j*3+l];
      for (int t = 0; t < 5; ++t) {
        float acc = 0.f;
        for (int kl = 0; kl < 9; ++kl) acc += T[kl] * m2[t*9+kl];
        o[160 + p++] = acc;
      }
    }
  p = 0;                                                    // contr_7 @235
  for (int r = 0; r < 5; ++r)
    for (int s = 0; s < 5; ++s)
      for (int t = 0; t < 5; ++t) {
        float acc = 0.f;
        for (int i = 0; i < 3; ++i)
          for (int j = 0; j < 3; ++j)
            for (int k = 0; k < 3; ++k)
              acc += m3[r*27+i*9+j*3+k] * m2[s*9+i*3+j] * m1[t*3+k];
        o[235 + p++] = acc;
      }
}

extern "C" void kernel_launch(void* const* d_in, const int* in_sizes, int n_in,
                              void* d_out, int out_size, void* d_ws,
                              size_t ws_size, hipStream_t stream) {
  const float* dr_vec = (const float*)d_in[0];
  const int*   Z      = (const int*)d_in[1];
  const int*   nbr    = (const int*)d_in[2];
  const float* coeffs = (const float*)d_in[3];
  float* out = (float*)d_out;
  float* ws  = (float*)d_ws;

  int nWs = N_ATOMS * ACC_STRIDE;
  zero_ws_kernel<<<(nWs + 255) / 256, 256, 0, stream>>>(ws, nWs);

  edge_kernel<<<(N_EDGES + 127) / 128, 128, 0, stream>>>(dr_vec, Z, nbr,
                                                         coeffs, ws);

  expand_kernel<<<(N_ATOMS + 255) / 256, 256, 0, stream>>>(ws);

  int nWaves = (N_ATOMS + 2) / 3;              // 3 atoms per wave
  int gramBlocks = (nWaves + WAVES_PER_BLOCK - 1) / WAVES_PER_BLOCK;
  gram_kernel<<<gramBlocks, 256, 0, stream>>>(ws, out);

  contract_kernel<<<(N_ATOMS + 255) / 256, 256, 0, stream>>>(ws, out);
}